// NLC_WeightNet_11768210391251
// MI455X (gfx1250) — compile-verified
//
#include <hip/hip_runtime.h>
#include <hip/hip_bf16.h>
#include <math.h>
#include <stdint.h>

static constexpr int BATCH = 32;
static constexpr int CIN   = 64;
static constexpr int COUT  = 64;
static constexpr int HD    = 112;
static constexpr int WDIM  = 112;
static constexpr int HW    = HD * WDIM;      // 12544
static constexpr int MO    = 128;            // attention channels
static constexpr int KC    = 576;            // CIN * 3 * 3
static constexpr int XIS   = 66;             // padded i-stride for x tile (even, 66%64==2)

typedef __attribute__((ext_vector_type(2))) float v2f;
typedef __attribute__((ext_vector_type(8))) float v8f;

// ---------------------------------------------------------------- utilities
__global__ void k_zero(float* p, int n) {
    int t = blockIdx.x * blockDim.x + threadIdx.x;
    if (t < n) p[t] = 0.0f;
}

// channel mean, 4 positions per thread (float4 loads, coalesced over s)
__global__ void k_xbar(const float* __restrict__ x, float* __restrict__ xbar) {
    int t = blockIdx.x * blockDim.x + threadIdx.x;        // over BATCH*HW/4
    if (t >= BATCH * HW / 4) return;
    int b = t / (HW / 4), s4 = t % (HW / 4);
    const float4* xp = (const float4*)(x + (size_t)b * CIN * HW) + s4;
    float4 acc = {0.f, 0.f, 0.f, 0.f};
    #pragma unroll
    for (int c = 0; c < CIN; ++c) {
        float4 v = xp[(size_t)c * (HW / 4)];
        acc.x += v.x; acc.y += v.y; acc.z += v.z; acc.w += v.w;
    }
    float4* ob = (float4*)(xbar + (size_t)b * HW) + s4;
    float4 r = {acc.x * (1.0f / CIN), acc.y * (1.0f / CIN),
                acc.z * (1.0f / CIN), acc.w * (1.0f / CIN)};
    *ob = r;
}

__global__ void k_sigmoid(const float* __restrict__ ctx, float* __restrict__ xw, int n) {
    int t = blockIdx.x * blockDim.x + threadIdx.x;
    if (t < n) xw[t] = 1.0f / (1.0f + __expf(-ctx[t]));
}

// per-sample dynamic weights, K-permuted: k' = (kh*3+kw)*64 + i
// wdyn[(b*COUT+o)*KC + k'] = xw[b, 2o + (i>=32)] * fc_w[f] + fc_b[f],  f = o*576 + i*9 + khkw
__global__ void k_wdyn(const float* __restrict__ fcw, const float* __restrict__ fcb,
                       const float* __restrict__ xw, float* __restrict__ wdyn) {
    int t = blockIdx.x * blockDim.x + threadIdx.x;        // over BATCH*COUT*KC
    if (t >= BATCH * COUT * KC) return;
    int kp = t % KC; int bo = t / KC; int o = bo % COUT; int b = bo / COUT;
    int i = kp & 63, khkw = kp >> 6;
    int flat = o * KC + i * 9 + khkw;
    wdyn[t] = xw[b * MO + 2 * o + (i >= 32)] * fcw[flat] + fcb[flat];
}

// ------------------------------------------------- attention: logits + softmax + context
// One wave handles 16 spatial positions; logits(128x16) = W(128x64) x x(64x16) via WMMA f32.
__global__ __launch_bounds__(256) void k_attn(
    const float* __restrict__ x, const float* __restrict__ wm,
    const float* __restrict__ bm, const float* __restrict__ xbar,
    float* __restrict__ ctx) {
    __shared__ float sW[MO * XIS];   // W row-major, padded col stride 66
    __shared__ float sB[MO];
    int tid = threadIdx.x;
    for (int e = tid; e < MO * CIN; e += 256) {
        int m = e >> 6, c = e & 63;
        sW[m * XIS + c] = wm[e];
    }
    if (tid < MO) sB[tid] = bm[tid];
    __syncthreads();

    int wid = tid >> 5, lane = tid & 31;
    int chunk = blockIdx.x * 8 + wid;                 // BATCH * (HW/16) chunks total
    int b  = chunk / (HW / 16);
    int s0 = (chunk % (HW / 16)) * 16;
    int n = lane & 15, hi = lane >> 4;

    // B fragments: x(64x16), lane n holds column s0+n; K split per ISA layout
    const float* xp = x + (size_t)b * CIN * HW + s0 + n;
    v2f bx[16];
    #pragma unroll
    for (int k = 0; k < 16; ++k) {
        int c0 = 4 * k + 2 * hi;
        v2f t; t.x = xp[(size_t)c0 * HW]; t.y = xp[(size_t)(c0 + 1) * HW];
        bx[k] = t;
    }

    v8f acc[8] = {};
    #pragma unroll
    for (int t = 0; t < 8; ++t) {
        const float* wrow = &sW[(t * 16 + n) * XIS + 2 * hi];
        #pragma unroll
        for (int k = 0; k < 16; ++k) {
            v2f a = *(const v2f*)(wrow + 4 * k);
            acc[t] = __builtin_amdgcn_wmma_f32_16x16x4_f32(
                false, a, false, bx[k], (short)0, acc[t], false, false);
        }
    }

    // bias + softmax over all 128 m (this lane holds 64, partner lane^16 the rest)
    float mx = -3.402823e38f;
    #pragma unroll
    for (int t = 0; t < 8; ++t)
        #pragma unroll
        for (int r = 0; r < 8; ++r) {
            float v = acc[t][r] + sB[t * 16 + r + 8 * hi];
            acc[t][r] = v;
            mx = fmaxf(mx, v);
        }
    mx = fmaxf(mx, __shfl_xor(mx, 16, 32));
    float sum = 0.0f;
    #pragma unroll
    for (int t = 0; t < 8; ++t)
        #pragma unroll
        for (int r = 0; r < 8; ++r) {
            float e = __expf(acc[t][r] - mx);
            acc[t][r] = e;
            sum += e;
        }
    sum += __shfl_xor(sum, 16, 32);
    float scale = xbar[(size_t)b * HW + s0 + n] / sum;

    #pragma unroll
    for (int t = 0; t < 8; ++t)
        #pragma unroll
        for (int r = 0; r < 8; ++r) {
            float v = acc[t][r] * scale;
            v += __shfl_xor(v, 1, 32);
            v += __shfl_xor(v, 2, 32);
            v += __shfl_xor(v, 4, 32);
            v += __shfl_xor(v, 8, 32);
            if (n == 0) atomicAdd(&ctx[b * MO + t * 16 + r + 8 * hi], v);
        }
}

// ------------------------------------------------- dynamic per-sample conv (implicit GEMM)
// block: 256 thr = 8 waves, covers (b, 32 out channels, 8 rows, 16 cols).
// x tile is pulled memory->LDS with GLOBAL_LOAD_ASYNC_TO_LDS_B32 (ASYNCcnt), halo zeros
// via plain ds stores. A-fragments stream from precomputed wdyn in global (imm-offset b64);
// B-fragments from LDS x tile [hh][ww][i]. Two o-tiles share every B fragment.
__global__ __launch_bounds__(256) void k_conv(
    const float* __restrict__ x, const float* __restrict__ wdyn,
    float* __restrict__ out) {
    __shared__ float sX[10 * 18 * XIS];     // rows hbase-1..hbase+8, cols wbase-1..wbase+16
    int tid = threadIdx.x;
    int wc = blockIdx.x % 7, hg = blockIdx.x / 7;     // 7 x 14
    int obase = blockIdx.y * 32;                      // 2 o-tiles of 16
    int b = blockIdx.z;
    int hbase = hg * 8, wbase = wc * 16;

    const float* xb = x + (size_t)b * CIN * HW;
    for (int e = tid; e < 64 * 10 * 18; e += 256) {
        int ww = e % 18; int t2 = e / 18; int hh = t2 % 10; int i = t2 / 10;
        int gh = hbase - 1 + hh, gw = wbase - 1 + ww;
        float* dst = &sX[(hh * 18 + ww) * XIS + i];
        if ((unsigned)gh < (unsigned)HD && (unsigned)gw < (unsigned)WDIM) {
            const float* src = &xb[(size_t)i * HW + gh * WDIM + gw];
            unsigned ldsoff = (unsigned)(uintptr_t)dst;   // wave-relative LDS address
            asm volatile("global_load_async_to_lds_b32 %0, %1, off"
                         :: "v"(ldsoff), "v"(src) : "memory");
        } else {
            *dst = 0.0f;                                   // zero halo
        }
    }
    asm volatile("s_wait_asynccnt 0x0" ::: "memory");
    __syncthreads();

    int wid = tid >> 5, lane = tid & 31;
    int n = lane & 15, hi = lane >> 4;
    int h = hbase + wid;

    const float* wrow0 = wdyn + (size_t)(b * COUT + obase + n) * KC + 2 * hi;
    const float* wrow1 = wrow0 + (size_t)16 * KC;

    v8f acc0 = {}, acc1 = {};
    #pragma unroll
    for (int khkw = 0; khkw < 9; ++khkw) {
        int kh = khkw / 3, kw = khkw - 3 * kh;
        const float* xrow = &sX[((wid + kh) * 18 + n + kw) * XIS + 2 * hi];
        #pragma unroll
        for (int i0 = 0; i0 < 64; i0 += 4) {
            int kp = khkw * 64 + i0;
            v2f a0 = *(const v2f*)(wrow0 + kp);
            v2f a1 = *(const v2f*)(wrow1 + kp);
            v2f bb = *(const v2f*)(xrow + i0);
            acc0 = __builtin_amdgcn_wmma_f32_16x16x4_f32(
                false, a0, false, bb, (short)0, acc0, false, false);
            acc1 = __builtin_amdgcn_wmma_f32_16x16x4_f32(
                false, a1, false, bb, (short)0, acc1, false, false);
        }
    }

    float* op0 = out + (((size_t)b * COUT + obase) * HD + h) * WDIM + wbase + n;
    #pragma unroll
    for (int r = 0; r < 8; ++r) {
        int ml = r + 8 * hi;
        op0[(size_t)ml * HW] = acc0[r];
        op0[(size_t)(ml + 16) * HW] = acc1[r];
    }
}

// ---------------------------------------------------------------- launcher
extern "C" void kernel_launch(void* const* d_in, const int* in_sizes, int n_in,
                              void* d_out, int out_size, void* d_ws, size_t ws_size,
                              hipStream_t stream) {
    const float* x   = (const float*)d_in[0];
    const float* wm  = (const float*)d_in[1];
    const float* bm  = (const float*)d_in[2];
    const float* fcw = (const float*)d_in[3];
    const float* fcb = (const float*)d_in[4];
    float* out = (float*)d_out;

    float* xbar = (float*)d_ws;                       // BATCH*HW
    float* ctx  = xbar + (size_t)BATCH * HW;          // BATCH*MO
    float* xw   = ctx + BATCH * MO;                   // BATCH*MO
    float* wdyn = xw + BATCH * MO;                    // BATCH*COUT*KC (4.7 MB)

    hipLaunchKernelGGL(k_zero, dim3((BATCH * MO + 255) / 256), dim3(256), 0, stream,
                       ctx, BATCH * MO);
    hipLaunchKernelGGL(k_xbar, dim3((BATCH * HW / 4 + 255) / 256), dim3(256), 0, stream,
                       x, xbar);
    hipLaunchKernelGGL(k_attn, dim3(BATCH * (HW / 16) / 8), dim3(256), 0, stream,
                       x, wm, bm, xbar, ctx);
    hipLaunchKernelGGL(k_sigmoid, dim3((BATCH * MO + 255) / 256), dim3(256), 0, stream,
                       ctx, xw, BATCH * MO);
    hipLaunchKernelGGL(k_wdyn, dim3((BATCH * COUT * KC + 255) / 256), dim3(256), 0, stream,
                       fcw, fcb, xw, wdyn);
    hipLaunchKernelGGL(k_conv, dim3(98, 2, BATCH), dim3(256), 0, stream,
                       x, wdyn, out);
}